// KGCCL_90890097918065
// MI455X (gfx1250) — compile-verified
//
#include <hip/hip_runtime.h>
#include <hip/hip_bf16.h>
#include <math.h>

typedef __attribute__((ext_vector_type(2))) float v2f;
typedef __attribute__((ext_vector_type(8))) float v8f;

#define D_DIM 64

__device__ __forceinline__ float wave32_sum(float v) {
#pragma unroll
  for (int off = 16; off >= 1; off >>= 1) v += __shfl_xor(v, off, 32);
  return v;
}
// reductions across each 16-lane half (xor masks <= 8 stay inside the half)
__device__ __forceinline__ float half16_sum(float v) {
#pragma unroll
  for (int off = 8; off >= 1; off >>= 1) v += __shfl_xor(v, off, 32);
  return v;
}
__device__ __forceinline__ float half16_max(float v) {
#pragma unroll
  for (int off = 8; off >= 1; off >>= 1) v = fmaxf(v, __shfl_xor(v, off, 32));
  return v;
}

// Pass 1: per edge, att = ||h*r||^2 * ||t*r||^2 ; scatter-max into mbuf (int-bit
// max valid: att >= 0). One wave32 per edge, float2 per lane covers D=64.
__global__ void edge_att_max(const float* __restrict__ ent,
                             const int* __restrict__ head,
                             const int* __restrict__ tail,
                             const int* __restrict__ etype,
                             const float* __restrict__ weight,
                             float* __restrict__ att,
                             int* __restrict__ mbuf, int E) {
  int lane = threadIdx.x & 31;
  long wid = ((long)blockIdx.x * blockDim.x + threadIdx.x) >> 5;
  if (wid >= E) return;
  int h = head[wid], t = tail[wid], r = etype[wid] - 1;
  const float2* hp = reinterpret_cast<const float2*>(ent + (size_t)h * D_DIM);
  const float2* tp = reinterpret_cast<const float2*>(ent + (size_t)t * D_DIM);
  const float2* rp = reinterpret_cast<const float2*>(weight + (size_t)r * D_DIM);
  float2 hv = hp[lane], tv = tp[lane], rv = rp[lane];
  float hx = hv.x * rv.x, hy = hv.y * rv.y;
  float tx = tv.x * rv.x, ty = tv.y * rv.y;
  float hn2 = wave32_sum(hx * hx + hy * hy);
  float tn2 = wave32_sum(tx * tx + ty * ty);
  if (lane == 0) {
    float a = hn2 * tn2;
    att[wid] = a;
    atomicMax(&mbuf[h], __float_as_int(a));
  }
}

// Pass 2: ex = exp(att - m[head]); att buffer overwritten in place; sum per head.
__global__ void edge_exp_sum(float* __restrict__ att,
                             const int* __restrict__ head,
                             const int* __restrict__ mbuf,
                             float* __restrict__ sbuf, int E) {
  int e = blockIdx.x * blockDim.x + threadIdx.x;
  if (e >= E) return;
  int h = head[e];
  float ex = __expf(att[e] - __int_as_float(mbuf[h]));
  att[e] = ex;
  atomicAdd(&sbuf[h], ex);
}

// Pass 3: entity_agg[head] += (ex/s[head]) * (entity_emb[tail] * rel)
__global__ void edge_scatter(const float* __restrict__ ent,
                             const int* __restrict__ head,
                             const int* __restrict__ tail,
                             const int* __restrict__ etype,
                             const float* __restrict__ weight,
                             const float* __restrict__ ex,
                             const float* __restrict__ sbuf,
                             float* __restrict__ entity_agg, int E) {
  int lane = threadIdx.x & 31;
  long wid = ((long)blockIdx.x * blockDim.x + threadIdx.x) >> 5;
  if (wid >= E) return;
  int h = head[wid], t = tail[wid], r = etype[wid] - 1;
  float w = ex[wid] / sbuf[h];
  const float2* tp = reinterpret_cast<const float2*>(ent + (size_t)t * D_DIM);
  const float2* rp = reinterpret_cast<const float2*>(weight + (size_t)r * D_DIM);
  float2 tv = tp[lane], rv = rp[lane];
  float* dst = entity_agg + (size_t)h * D_DIM + lane * 2;
  atomicAdd(dst + 0, w * tv.x * rv.x);
  atomicAdd(dst + 1, w * tv.y * rv.y);
}

// Pass 4: COO SpMM: user_agg[row] += val * entity_emb[col]
__global__ void spmm_coo(const float* __restrict__ ent,
                         const int* __restrict__ row,
                         const int* __restrict__ col,
                         const float* __restrict__ val,
                         float* __restrict__ user_agg, int NNZ) {
  int lane = threadIdx.x & 31;
  long wid = ((long)blockIdx.x * blockDim.x + threadIdx.x) >> 5;
  if (wid >= NNZ) return;
  int rI = row[wid], c = col[wid];
  float v = val[wid];
  const float2* cp = reinterpret_cast<const float2*>(ent + (size_t)c * D_DIM);
  float2 cv = cp[lane];
  float* dst = user_agg + (size_t)rI * D_DIM + lane * 2;
  atomicAdd(dst + 0, v * cv.x);
  atomicAdd(dst + 1, v * cv.y);
}

// Pass 5: WMMA gating. One wave per 16-user tile.
// GEMM1: logits[16x32pad] = user_emb_tile[16x64] @ weight.T[64x32pad]
// softmax over 31 cols -> score ; LDS bounce C-layout -> A-layout
// GEMM2: corr[16x64] = score[16x32] @ weight[32pad x 64]
// user_agg = user_agg * (1 + corr)
__global__ void user_gemm(const float* __restrict__ user_emb,
                          const float* __restrict__ weight,
                          float* __restrict__ user_agg, int n_rel_m1) {
  __shared__ float w_lds[32 * 65];   // weight padded to 32 rows, stride 65
  __shared__ float sc_lds[16 * 33];  // score 16x32, stride 33

  int l = threadIdx.x;       // 0..31, one wave
  int row_in = l & 15;       // M row (A) or N col (B/C)
  int khalf = (l >> 4) * 2;  // K sub-pair select for A/B layout

  for (int idx = l; idx < 32 * 64; idx += 32) {
    int r = idx >> 6, d = idx & 63;
    w_lds[r * 65 + d] = (r < n_rel_m1) ? weight[r * 64 + d] : 0.0f;
  }
  __syncthreads();

  int tile = blockIdx.x;
  const float* urow = user_emb + ((size_t)tile * 16 + row_in) * 64;

  v8f accA = {0.f, 0.f, 0.f, 0.f, 0.f, 0.f, 0.f, 0.f};
  v8f accB = {0.f, 0.f, 0.f, 0.f, 0.f, 0.f, 0.f, 0.f};
#pragma unroll
  for (int k0 = 0; k0 < 64; k0 += 4) {
    int kk = k0 + khalf;
    v2f a;  a.x = urow[kk];                         a.y = urow[kk + 1];
    v2f bA; bA.x = w_lds[row_in * 65 + kk];         bA.y = w_lds[row_in * 65 + kk + 1];
    v2f bB; bB.x = w_lds[(16 + row_in) * 65 + kk];  bB.y = w_lds[(16 + row_in) * 65 + kk + 1];
    accA = __builtin_amdgcn_wmma_f32_16x16x4_f32(false, a, false, bA, (short)0, accA, false, false);
    accB = __builtin_amdgcn_wmma_f32_16x16x4_f32(false, a, false, bB, (short)0, accB, false, false);
  }

  // softmax over cols 0..30 per user row (col 31 is padding)
  bool padB = (16 + row_in) >= n_rel_m1;
  int lds_row = (l >> 4) * 8;  // upper half-wave holds rows 8..15
#pragma unroll
  for (int v = 0; v < 8; v++) {
    float v0 = accA[v];
    float v1 = padB ? -3.0e38f : accB[v];
    float mx = half16_max(fmaxf(v0, v1));
    float e0 = __expf(v0 - mx);
    float e1 = padB ? 0.0f : __expf(v1 - mx);
    float s = half16_sum(e0 + e1);
    int rr = v + lds_row;
    sc_lds[rr * 33 + row_in] = e0 / s;
    sc_lds[rr * 33 + 16 + row_in] = e1 / s;
  }
  __syncthreads();

  v8f c0 = {0.f,0.f,0.f,0.f,0.f,0.f,0.f,0.f};
  v8f c1 = c0, c2 = c0, c3 = c0;
#pragma unroll
  for (int k0 = 0; k0 < 32; k0 += 4) {
    int kk = k0 + khalf;
    v2f a; a.x = sc_lds[row_in * 33 + kk]; a.y = sc_lds[row_in * 33 + kk + 1];
    v2f b;
    b.x = w_lds[kk * 65 + row_in];       b.y = w_lds[(kk + 1) * 65 + row_in];
    c0 = __builtin_amdgcn_wmma_f32_16x16x4_f32(false, a, false, b, (short)0, c0, false, false);
    b.x = w_lds[kk * 65 + 16 + row_in];  b.y = w_lds[(kk + 1) * 65 + 16 + row_in];
    c1 = __builtin_amdgcn_wmma_f32_16x16x4_f32(false, a, false, b, (short)0, c1, false, false);
    b.x = w_lds[kk * 65 + 32 + row_in];  b.y = w_lds[(kk + 1) * 65 + 32 + row_in];
    c2 = __builtin_amdgcn_wmma_f32_16x16x4_f32(false, a, false, b, (short)0, c2, false, false);
    b.x = w_lds[kk * 65 + 48 + row_in];  b.y = w_lds[(kk + 1) * 65 + 48 + row_in];
    c3 = __builtin_amdgcn_wmma_f32_16x16x4_f32(false, a, false, b, (short)0, c3, false, false);
  }

#pragma unroll
  for (int v = 0; v < 8; v++) {
    int rr = v + lds_row;
    float* up = user_agg + ((size_t)tile * 16 + rr) * 64 + row_in;
    float u0 = up[0];  up[0]  = u0 + c0[v] * u0;
    float u1 = up[16]; up[16] = u1 + c1[v] * u1;
    float u2 = up[32]; up[32] = u2 + c2[v] * u2;
    float u3 = up[48]; up[48] = u3 + c3[v] * u3;
  }
}

extern "C" void kernel_launch(void* const* d_in, const int* in_sizes, int n_in,
                              void* d_out, int out_size, void* d_ws, size_t ws_size,
                              hipStream_t stream) {
  (void)n_in; (void)ws_size;
  const float* entity_emb = (const float*)d_in[0];
  const float* user_emb   = (const float*)d_in[1];
  const int*   edge_index = (const int*)d_in[2];
  const int*   edge_type  = (const int*)d_in[3];
  const int*   irow       = (const int*)d_in[4];
  const int*   icol       = (const int*)d_in[5];
  const float* ival       = (const float*)d_in[6];
  const float* weight     = (const float*)d_in[7];

  int n_ent    = in_sizes[0] / D_DIM;
  int n_usr    = in_sizes[1] / D_DIM;
  int E        = in_sizes[3];
  int NNZ      = in_sizes[6];
  int n_rel_m1 = in_sizes[7] / D_DIM;  // 31

  const int* head = edge_index;
  const int* tail = edge_index + E;

  float* entity_agg = (float*)d_out;
  float* user_agg   = entity_agg + (size_t)n_ent * D_DIM;

  // workspace: att/ex [E] | mbuf [n_ent] | sbuf [n_ent]  (~4.8 MB)
  float* att  = (float*)d_ws;
  int*   mbuf = (int*)(att + E);
  float* sbuf = (float*)(mbuf + n_ent);

  hipMemsetAsync(d_out, 0, (size_t)out_size * sizeof(float), stream);
  hipMemsetAsync(mbuf, 0, (size_t)n_ent * sizeof(int), stream);    // 0.0f bits
  hipMemsetAsync(sbuf, 0, (size_t)n_ent * sizeof(float), stream);

  int gridE = (E + 7) / 8;  // 8 waves (256 threads) per block, wave per edge
  edge_att_max<<<gridE, 256, 0, stream>>>(entity_emb, head, tail, edge_type,
                                          weight, att, mbuf, E);
  edge_exp_sum<<<(E + 255) / 256, 256, 0, stream>>>(att, head, mbuf, sbuf, E);
  edge_scatter<<<gridE, 256, 0, stream>>>(entity_emb, head, tail, edge_type,
                                          weight, att, sbuf, entity_agg, E);
  spmm_coo<<<(NNZ + 7) / 8, 256, 0, stream>>>(entity_emb, irow, icol, ival,
                                              user_agg, NNZ);
  user_gemm<<<n_usr / 16, 32, 0, stream>>>(user_emb, weight, user_agg, n_rel_m1);
}